// TwoLayerSimpleLightGCN_5265629905489
// MI455X (gfx1250) — compile-verified
//
#include <hip/hip_runtime.h>
#include <cstdint>
#include <cstddef>

// ---------------------------------------------------------------------------
// TwoLayerSimpleLightGCN for MI455X (gfx1250)
//
// Roofline: ~0.8 GFLOP vs ~3.5 GB of gather/scatter traffic -> entirely
// memory/atomic bound; WMMA does not apply (no dense matmul anywhere).
// MI455X-specific exploitation:
//   * 192 MB L2: full working set (emb tables + accumulators ~155 MB) is
//     L2-resident, so random 256B row gathers and f32 atomic scatters are
//     served by L2 atomic units, not HBM.
//   * Explicit single-instruction device-scope f32 atomics:
//       global_atomic_add_f32 ... scope:SCOPE_DEV   (no CAS-loop fallback)
//   * wave32: 16 lanes x float4 cover one 64-float row; 2 edges per wave.
//   * CDNA5 async LDS path: the streaming "finish" pass stages its B128
//     reads via global_load_async_to_lds_b128 + s_wait_asynccnt (ASYNCcnt).
// ---------------------------------------------------------------------------

#define EMB 64  // EMB_DIM in the reference

typedef __attribute__((address_space(3))) const float4 lds_cfloat4;

// Guaranteed hardware f32 atomic add, no-return form (STOREcnt-tracked),
// resolved at the device coherence point (L2 atomic units).
__device__ __forceinline__ void global_fadd_dev(float* p, float v) {
  asm volatile("global_atomic_add_f32 %0, %1, off scope:SCOPE_DEV"
               :: "v"((unsigned long long)(uintptr_t)p), "v"(v)
               : "memory");
}

// ---------------------------------------------------------------------------
// Edge-parallel scatter: for each edge e, adds usrc[src[e]] into isum[dst[e]]
// and isrc[dst[e]] into usum[src[e]] (both directions of one layer in one
// pass, so src[e]/dst[e] are read once). 16 lanes per edge, float4 per lane.
// Optionally counts degrees (only needed on the first layer).
// ---------------------------------------------------------------------------
__global__ void edge_scatter_kernel(const float* __restrict__ usrc,
                                    const float* __restrict__ isrc,
                                    const int* __restrict__ src,
                                    const int* __restrict__ dst,
                                    float* __restrict__ usum,
                                    float* __restrict__ isum,
                                    float* __restrict__ deg_u,
                                    float* __restrict__ deg_i,
                                    int count_deg, int nedges) {
  const long long tid = (long long)blockIdx.x * blockDim.x + threadIdx.x;
  const int e    = (int)(tid >> 4);
  const int lane = (int)(tid & 15);
  if (e >= nedges) return;

  if (lane == 0) {
    // keep the (tiny) index stream ahead of us (near-scope prefetch)
    __builtin_prefetch(src + e + 4096, 0, 3);
    __builtin_prefetch(dst + e + 4096, 0, 3);
  }

  const int s = src[e];
  const int d = dst[e];

  const float4 uv = *(const float4*)(usrc + (size_t)s * EMB + lane * 4);  // 256B coalesced
  const float4 iv = *(const float4*)(isrc + (size_t)d * EMB + lane * 4);

  float* ip = isum + (size_t)d * EMB + lane * 4;
  float* up = usum + (size_t)s * EMB + lane * 4;

  // f32 atomics resolved at L2 (tables fit in 192MB L2)
  global_fadd_dev(ip + 0, uv.x);
  global_fadd_dev(ip + 1, uv.y);
  global_fadd_dev(ip + 2, uv.z);
  global_fadd_dev(ip + 3, uv.w);

  global_fadd_dev(up + 0, iv.x);
  global_fadd_dev(up + 1, iv.y);
  global_fadd_dev(up + 2, iv.z);
  global_fadd_dev(up + 3, iv.w);

  if (count_deg && lane == 0) {
    global_fadd_dev(deg_u + s, 1.0f);
    global_fadd_dev(deg_i + d, 1.0f);
  }
}

// ---------------------------------------------------------------------------
// finish: out = sum / max(deg,1)  (== reference where(cnt>0, s/cnt, 0) since
// sum==0 whenever deg==0), and re-zero `sum` so the buffer can be reused for
// layer 2 without an extra memset pass.
// The streaming read of `sum` is staged through LDS with the CDNA5 async
// load path (ASYNCcnt): global_load_async_to_lds_b128 + s_wait_asynccnt.
// Each lane reads back only its own staged slot, so a per-wave counter wait
// is sufficient (no barrier needed).
// ---------------------------------------------------------------------------
__global__ void finish_mean_kernel(float* __restrict__ sum,
                                   const float* __restrict__ deg,
                                   float* __restrict__ out,
                                   int nnodes) {
  __shared__ float4 stage[256];

  const int tid    = blockIdx.x * blockDim.x + threadIdx.x;  // one float4 each
  const int total4 = nnodes * (EMB / 4);
  if (tid >= total4) return;

  lds_cfloat4* sp = (lds_cfloat4*)(&stage[threadIdx.x]);
  const uint32_t lds_off = (uint32_t)(size_t)sp;          // raw LDS byte offset
  const uint32_t voff    = (uint32_t)tid * 16u;           // byte offset into sum
  const unsigned long long base = (unsigned long long)(uintptr_t)sum;

  // CDNA5 async global->LDS copy (GVS mode: saddr base + 32-bit vaddr offset)
  asm volatile("global_load_async_to_lds_b128 %0, %1, %2"
               :: "v"(lds_off), "v"(voff), "s"(base)
               : "memory");
  __builtin_amdgcn_s_wait_asynccnt(0);
  asm volatile("" ::: "memory");  // keep the LDS read below the wait

  const float4 s4 = stage[threadIdx.x];

  const int   node  = tid >> 4;
  const float denom = fmaxf(deg[node], 1.0f);

  float4 r;
  r.x = s4.x / denom;
  r.y = s4.y / denom;
  r.z = s4.z / denom;
  r.w = s4.w / denom;
  *(float4*)(out + (size_t)tid * 4) = r;

  // consume-and-clear so layer 2 starts from zeroed accumulators
  *(float4*)(sum + (size_t)tid * 4) = make_float4(0.f, 0.f, 0.f, 0.f);
}

// ---------------------------------------------------------------------------
// final: out = (x0 + x1 + sum2/max(deg,1)) / 3, where `out` currently holds x1.
// Divide by exactly 3.0f to match the reference's (a+b+c)/3.0 rounding.
// ---------------------------------------------------------------------------
__global__ void final_combine_kernel(const float* __restrict__ x0,
                                     const float* __restrict__ sum2,
                                     const float* __restrict__ deg,
                                     float* __restrict__ out,
                                     int nnodes) {
  const int tid    = blockIdx.x * blockDim.x + threadIdx.x;  // one float4 each
  const int total4 = nnodes * (EMB / 4);
  if (tid >= total4) return;

  const int   node  = tid >> 4;
  const float denom = fmaxf(deg[node], 1.0f);

  const float4 a = *(const float4*)(x0   + (size_t)tid * 4);
  const float4 b = *(const float4*)(out  + (size_t)tid * 4);
  const float4 c = *(const float4*)(sum2 + (size_t)tid * 4);

  float4 r;
  r.x = (a.x + b.x + c.x / denom) / 3.0f;
  r.y = (a.y + b.y + c.y / denom) / 3.0f;
  r.z = (a.z + b.z + c.z / denom) / 3.0f;
  r.w = (a.w + b.w + c.w / denom) / 3.0f;
  *(float4*)(out + (size_t)tid * 4) = r;
}

// ---------------------------------------------------------------------------
// Host-side orchestration (graph-capture safe: only kernel launches and one
// hipMemsetAsync on `stream`).
//
// Workspace layout (floats):  [usum: Nu*64][isum: Ni*64][deg_u: Nu][deg_i: Ni]
//   = 78 MB for the reference sizes. d_out holds u1/i1 between the layers,
// so no separate x1 buffers are needed.
// ---------------------------------------------------------------------------
extern "C" void kernel_launch(void* const* d_in, const int* in_sizes, int n_in,
                              void* d_out, int out_size, void* d_ws, size_t ws_size,
                              hipStream_t stream) {
  const float* u0  = (const float*)d_in[0];
  const float* i0  = (const float*)d_in[1];
  const int*   src = (const int*)d_in[2];
  const int*   dst = (const int*)d_in[3];

  const int n_users = in_sizes[0] / EMB;
  const int n_items = in_sizes[1] / EMB;
  const int nedges  = in_sizes[2];

  float* out_user = (float*)d_out;
  float* out_item = out_user + (size_t)n_users * EMB;

  float* usum  = (float*)d_ws;
  float* isum  = usum + (size_t)n_users * EMB;
  float* deg_u = isum + (size_t)n_items * EMB;
  float* deg_i = deg_u + n_users;
  const size_t ws_floats =
      (size_t)(n_users + n_items) * EMB + (size_t)n_users + (size_t)n_items;

  // ws is poisoned by the harness; zero accumulators + degrees every call.
  hipMemsetAsync(d_ws, 0, ws_floats * sizeof(float), stream);

  const int B = 256;
  const long long ethreads = (long long)nedges * 16;
  const int eblocks = (int)((ethreads + B - 1) / B);
  const int ublocks = (n_users * (EMB / 4) + B - 1) / B;
  const int iblocks = (n_items * (EMB / 4) + B - 1) / B;

  // ----- layer 1: i1 = mean_dst(u0[src]), u1 = mean_src(i0[dst]) -----
  edge_scatter_kernel<<<eblocks, B, 0, stream>>>(u0, i0, src, dst,
                                                 usum, isum, deg_u, deg_i,
                                                 /*count_deg=*/1, nedges);
  finish_mean_kernel<<<ublocks, B, 0, stream>>>(usum, deg_u, out_user, n_users);
  finish_mean_kernel<<<iblocks, B, 0, stream>>>(isum, deg_i, out_item, n_items);

  // ----- layer 2: gathers x1 directly from d_out (sums were re-zeroed) -----
  edge_scatter_kernel<<<eblocks, B, 0, stream>>>(out_user, out_item, src, dst,
                                                 usum, isum, deg_u, deg_i,
                                                 /*count_deg=*/0, nedges);

  // ----- final: out = (x0 + x1 + x2) / 3, in place over x1 -----
  final_combine_kernel<<<ublocks, B, 0, stream>>>(u0, usum, deg_u, out_user, n_users);
  final_combine_kernel<<<iblocks, B, 0, stream>>>(i0, isum, deg_i, out_item, n_items);
}